// AttentionMechanism_6811818132381
// MI455X (gfx1250) — compile-verified
//
#include <hip/hip_runtime.h>
#include <hip/hip_bf16.h>
#include <stdint.h>

// Problem constants (from reference)
#define BB 2
#define QQ 512
#define KK 1024
#define DD 256   // QD == KD
#define HH 128
#define TK 64    // pk rows staged in LDS per tile

typedef float v2f __attribute__((ext_vector_type(2)));
typedef float v8f __attribute__((ext_vector_type(8)));

// ---------------------------------------------------------------------------
// Fast tanh: prefer hardware v_tanh_f32 if the builtin exists, else one
// v_exp_f32: tanh(x) = 1 - 2/(exp2(2x*log2e)+1). Clamp so exp2 never overflows.
// ---------------------------------------------------------------------------
__device__ __forceinline__ float fast_tanh(float x) {
#if __has_builtin(__builtin_amdgcn_tanhf)
    return __builtin_amdgcn_tanhf(x);
#else
    float xc = fminf(fmaxf(x, -10.0f), 10.0f);
    float t  = __builtin_amdgcn_exp2f(xc * 2.8853900817779268f); // 2*log2(e)
    return 1.0f - 2.0f / (t + 1.0f);
#endif
}

// Async global -> LDS copy of one 16-byte chunk per lane (gfx1250).
// Tracked by ASYNCcnt; caller must s_wait_asynccnt + barrier before reading.
__device__ __forceinline__ void async_copy_b128(void* lds_dst, const void* gsrc) {
    uint32_t lds = (uint32_t)(uintptr_t)lds_dst;   // LDS byte address
    uint64_t ga  = (uint64_t)(uintptr_t)gsrc;      // global address
    asm volatile("global_load_async_to_lds_b128 %0, %1, off"
                 :: "v"(lds), "v"(ga) : "memory");
}

__device__ __forceinline__ void wait_asynccnt0() {
#if __has_builtin(__builtin_amdgcn_s_wait_asynccnt)
    __builtin_amdgcn_s_wait_asynccnt(0);
#else
    asm volatile("s_wait_asynccnt 0x0" ::: "memory");
#endif
}

// ---------------------------------------------------------------------------
// Generic f32 WMMA GEMM: C[M x N] = A[M x Kdim] * B  (+batch strides)
//   BT = false : B is row-major [Kdim x N],  B(k,n) = Bm[k*ldb + n]
//   BT = true  : B is row-major [N x Kdim],  B(k,n) = Bm[n*ldb + k]  (i.e. A*B^T)
// One wave32 computes one 16x16 tile with V_WMMA_F32_16X16X4_F32.
// All dims are multiples of 16; Kdim multiple of 4. EXEC stays all-ones.
// ---------------------------------------------------------------------------
template <bool BT>
__global__ void wmma_gemm_f32_kernel(const float* __restrict__ A,
                                     const float* __restrict__ Bm,
                                     float* __restrict__ C,
                                     int Kdim, int lda, int ldb, int ldc,
                                     long sA, long sB, long sC) {
    const int lane  = threadIdx.x;                       // 0..31
    const int tileN = blockIdx.x;
    const int tileM = blockIdx.y * blockDim.y + threadIdx.y;
    const int batch = blockIdx.z;

    A  += (long)batch * sA;
    Bm += (long)batch * sB;
    C  += (long)batch * sC;

    const int mn   = lane & 15;            // M index (A) / N index (B, C/D)
    const int koff = (lane >> 4) * 2;      // 0 or 2
    const int row  = tileM * 16 + mn;
    const int col  = tileN * 16 + mn;

    v8f acc = {};

#pragma unroll 8
    for (int k0 = 0; k0 < Kdim; k0 += 4) {
        v2f a, b;
        const float* ap = A + (long)row * lda + (k0 + koff);
        a.x = ap[0];
        a.y = ap[1];
        if (BT) {
            const float* bp = Bm + (long)col * ldb + (k0 + koff);
            b.x = bp[0];
            b.y = bp[1];
        } else {
            b.x = Bm[(long)(k0 + koff) * ldb + col];
            b.y = Bm[(long)(k0 + koff + 1) * ldb + col];
        }
        acc = __builtin_amdgcn_wmma_f32_16x16x4_f32(
            /*neg_a=*/false, a, /*neg_b=*/false, b,
            /*c_mod=*/(short)0, acc, /*reuse_a=*/false, /*reuse_b=*/false);
    }

    const int rbase = tileM * 16 + (lane >> 4) * 8;
#pragma unroll
    for (int i = 0; i < 8; ++i) {
        C[(long)(rbase + i) * ldc + col] = acc[i];
    }
}

// ---------------------------------------------------------------------------
// energy + mask + softmax: one block (256 threads) per (b,q) row.
//   energy[k] = sum_h tanh(pq[b,q,h] + pk[b,k,h]) * We[h]
// pk staged in LDS in 64-row tiles via global_load_async_to_lds_b128.
// 4 adjacent lanes split the 128-wide h reduction per k (32 h each), combined
// with two wave32 shfl_xor steps. Then masked softmax over k.
// ---------------------------------------------------------------------------
__global__ void energy_softmax_kernel(const float* __restrict__ pq,
                                      const float* __restrict__ pk,
                                      const int* __restrict__ mask,
                                      const float* __restrict__ We,
                                      float* __restrict__ attn) {
    const int q = blockIdx.x;
    const int b = blockIdx.y;
    const int t = threadIdx.x;                 // 0..255

    __shared__ float s_pk[TK * HH];            // 32 KB tile
    __shared__ float s_pq[HH];
    __shared__ float s_we[HH];
    __shared__ float s_e[KK];
    __shared__ float s_red[256];

    if (t < HH) {
        s_pq[t] = pq[((long)b * QQ + q) * HH + t];
        s_we[t] = We[t];
    }

    const float* pkb  = pk + (long)b * KK * HH;
    const int*   mrow = mask + ((long)b * QQ + q) * KK;
    float*       arow = attn + ((long)b * QQ + q) * KK;

    const int kl  = t >> 2;          // k within tile: 0..63
    const int hq  = (t & 3) * 32;    // this lane's h quarter
    __syncthreads();

    // ---- energy: 16 tiles of 64 k-rows -----------------------------------
    for (int kt = 0; kt < KK; kt += TK) {
        // Cooperative async copy: TK*HH floats = 2048 float4 / 256 threads.
        const float* src = pkb + (long)kt * HH;
#pragma unroll
        for (int i = 0; i < (TK * HH) / (256 * 4); ++i) {
            const int idx4 = (t + i * 256) * 4;          // float index
            async_copy_b128(s_pk + idx4, src + idx4);
        }
        // Prefetch next tile into L2 while this one lands in LDS.
        if (kt + TK < KK) {
            __builtin_prefetch(pkb + (long)(kt + TK) * HH + t * 32, 0, 0);
        }
        wait_asynccnt0();
        __syncthreads();

        const float* row = s_pk + kl * HH + hq;
        const float* wp  = s_we + hq;
        const float* qp  = s_pq + hq;
        float e = 0.0f;
#pragma unroll 8
        for (int h = 0; h < 32; ++h) {
            e = fmaf(fast_tanh(qp[h] + row[h]), wp[h], e);
        }
        // combine the 4 lanes covering this k (lanes l, l^1, l^2 in-wave)
        e += __shfl_xor(e, 1, 32);
        e += __shfl_xor(e, 2, 32);
        if ((t & 3) == 0) {
            const int k = kt + kl;
            s_e[k] = (mrow[k] == 0) ? -1.0e9f : e;
        }
        __syncthreads();
    }

    // ---- row max ----------------------------------------------------------
    float m = -3.0e38f;
    for (int k = t; k < KK; k += 256) m = fmaxf(m, s_e[k]);
    s_red[t] = m;
    __syncthreads();
#pragma unroll
    for (int s = 128; s > 0; s >>= 1) {
        if (t < s) s_red[t] = fmaxf(s_red[t], s_red[t + s]);
        __syncthreads();
    }
    const float mx = s_red[0];
    __syncthreads();

    // ---- exp + sum --------------------------------------------------------
    float sum = 0.0f;
    for (int k = t; k < KK; k += 256) {
        float e = __builtin_amdgcn_exp2f((s_e[k] - mx) * 1.4426950408889634f);
        s_e[k] = e;
        sum += e;
    }
    s_red[t] = sum;
    __syncthreads();
#pragma unroll
    for (int s = 128; s > 0; s >>= 1) {
        if (t < s) s_red[t] += s_red[t + s];
        __syncthreads();
    }
    const float inv = 1.0f / s_red[0];

    // ---- normalize + write ------------------------------------------------
    for (int k = t; k < KK; k += 256) {
        arow[k] = s_e[k] * inv;
    }
}

// ---------------------------------------------------------------------------
// launch
// ---------------------------------------------------------------------------
extern "C" void kernel_launch(void* const* d_in, const int* in_sizes, int n_in,
                              void* d_out, int out_size, void* d_ws, size_t ws_size,
                              hipStream_t stream) {
    const float* queries = (const float*)d_in[0];  // (B,Q,QD)
    const float* keys    = (const float*)d_in[1];  // (B,K,KD)
    const int*   mask    = (const int*)d_in[2];    // (B,Q,K)
    const float* Wq      = (const float*)d_in[3];  // (H,QD)
    const float* Wk      = (const float*)d_in[4];  // (H,KD)
    const float* We      = (const float*)d_in[5];  // (H)

    float* context = (float*)d_out;                        // (B,Q,QD)
    float* attn    = (float*)d_out + (long)BB * QQ * DD;   // (B,Q,K)

    float* pq = (float*)d_ws;                              // (B,Q,H)
    float* pk = pq + (long)BB * QQ * HH;                   // (B,K,H)

    dim3 blk(32, 4);

    // pq = queries @ Wq^T : M=Q, N=H, Kdim=QD, B transposed
    wmma_gemm_f32_kernel<true><<<dim3(HH / 16, QQ / 64, BB), blk, 0, stream>>>(
        queries, Wq, pq, /*Kdim=*/DD, /*lda=*/DD, /*ldb=*/DD, /*ldc=*/HH,
        (long)QQ * DD, 0L, (long)QQ * HH);

    // pk = keys @ Wk^T : M=K, N=H, Kdim=KD, B transposed
    wmma_gemm_f32_kernel<true><<<dim3(HH / 16, KK / 64, BB), blk, 0, stream>>>(
        keys, Wk, pk, /*Kdim=*/DD, /*lda=*/DD, /*ldb=*/DD, /*ldc=*/HH,
        (long)KK * DD, 0L, (long)KK * HH);

    // energy + masked softmax -> attn (second half of d_out)
    energy_softmax_kernel<<<dim3(QQ, BB), dim3(256), 0, stream>>>(
        pq, pk, mask, We, attn);

    // context = attn @ keys : M=Q, N=KD, Kdim=K, B row-major
    wmma_gemm_f32_kernel<false><<<dim3(DD / 16, QQ / 64, BB), blk, 0, stream>>>(
        attn, keys, context, /*Kdim=*/KK, /*lda=*/KK, /*ldb=*/DD, /*ldc=*/DD,
        (long)QQ * KK, (long)KK * DD, (long)QQ * DD);
}